// CoreRNNFW_12421045420533
// MI455X (gfx1250) — compile-verified
//
#include <hip/hip_runtime.h>
#include <hip/hip_bf16.h>

// ---------------------------------------------------------------------------
// Fast-weight RNN forward on MI455X (gfx1250, wave32).
//
//  * A_t = lambda*A_{t-1} + eta*h h^T is never materialized: per-batch history
//    H = {h_0..h_{t-1}} lives in LDS (48*512*4B = 96KB of CDNA5's 320KB WGP
//    LDS) and A.h = eta * sum_s lam^{t-1-s} (h_s.h) h_s.
//  * One workgroup per batch element (48 WGs), 256 threads (8 wave32s).
//  * base = W_h h + W_g z + b_h uses V_WMMA_F32_16X16X4_F32 (native fp32
//    WMMA): A = 16-row W tile, B column 0 = h[k..k+3] (other columns zero),
//    so D column 0 (lanes 0/16) accumulates the exact fp32 matvec.
//  * W streaming is L2-latency-bound -> b128 loads (full A-tile per load),
//    8 loads/8 WMMAs in flight per unrolled body, dual accumulator chains,
//    and global_prefetch_b8 three blocks (96 floats) ahead.
// ---------------------------------------------------------------------------

typedef __attribute__((ext_vector_type(2))) float v2f;
typedef __attribute__((ext_vector_type(8))) float v8f;

#define T_STEPS 48
#define BATCH   48
#define DG      256
#define DH      512
#define LAMBDA  0.95f
#define ETA     0.5f

__device__ __forceinline__ float wave_sum(float v) {
#pragma unroll
    for (int o = 16; o > 0; o >>= 1) v += __shfl_xor(v, o, 32);
    return v;
}

// LayerNorm (eps=1e-5) + ReLU over 512 elems; thread owns i=tid and i=tid+256.
__device__ __forceinline__ void ln_relu(float x0, float x1,
                                        const float* __restrict__ g,
                                        const float* __restrict__ bt,
                                        float* __restrict__ out,
                                        float* __restrict__ red,
                                        int tid) {
    const int lane = tid & 31, wave = tid >> 5;
    float s1 = x0 + x1;
    float s2 = x0 * x0 + x1 * x1;
    s1 = wave_sum(s1);
    s2 = wave_sum(s2);
    __syncthreads();                       // protect `red` reuse
    if (lane == 0) { red[wave] = s1; red[8 + wave] = s2; }
    __syncthreads();
    float t1 = 0.f, t2 = 0.f;
#pragma unroll
    for (int w = 0; w < 8; ++w) { t1 += red[w]; t2 += red[8 + w]; }
    const float mu  = t1 * (1.0f / DH);
    const float var = t2 * (1.0f / DH) - mu * mu;
    const float inv = rsqrtf(var + 1e-5f);
    const float y0 = (x0 - mu) * inv * g[tid]       + bt[tid];
    const float y1 = (x1 - mu) * inv * g[tid + 256] + bt[tid + 256];
    out[tid]       = fmaxf(y0, 0.f);
    out[tid + 256] = fmaxf(y1, 0.f);
}

__global__ void init_out_kernel(float* __restrict__ out) {
    if (threadIdx.x == 0) { out[0] = 0.f; out[1] = 0.f; }
}

__global__ void __launch_bounds__(256, 1)
fastweight_rnn_kernel(const float* __restrict__ z_seq,
                      const float* __restrict__ clean,
                      const float* __restrict__ Wh,
                      const float* __restrict__ Wg,
                      const float* __restrict__ bh,
                      const float* __restrict__ gamma,
                      const float* __restrict__ beta,
                      const float* __restrict__ Whead,
                      const float* __restrict__ bhead,
                      float* __restrict__ out) {
    __shared__ float sh_hist[T_STEPS * DH];   // 96 KB history of h_t
    __shared__ float sh_h[DH];
    __shared__ float sh_base[DH];
    __shared__ float sh_z[DG];
    __shared__ float sh_g[DH];
    __shared__ float sh_bt[DH];
    __shared__ float sh_bh[DH];
    __shared__ float sh_d[64];                // decayed dot coefficients
    __shared__ float sh_red[32];              // block reduction scratch
    __shared__ float sh_pow[64];              // lambda^p table

    const int tid  = threadIdx.x;             // 0..255
    const int lane = tid & 31;                // wave32
    const int wave = tid >> 5;                // 8 waves
    const int b    = blockIdx.x;              // batch element

    // Preload small params into LDS; h0 = 0.
    sh_g[tid]  = gamma[tid];  sh_g[tid + 256]  = gamma[tid + 256];
    sh_bt[tid] = beta[tid];   sh_bt[tid + 256] = beta[tid + 256];
    sh_bh[tid] = bh[tid];     sh_bh[tid + 256] = bh[tid + 256];
    sh_h[tid]  = 0.f;         sh_h[tid + 256]  = 0.f;
    if (tid == 0) {
        float p = 1.f;
        for (int i = 0; i < T_STEPS; ++i) { sh_pow[i] = p; p *= LAMBDA; }
    }
    __syncthreads();

    // WMMA f32 16x16x4 A-operand layout: VGPR0: K=0 (lanes 0-15) / K=2
    // (lanes 16-31); VGPR1: K=1 / K=3. Lane L and L+16 own the same M row.
    const int  khalf = (lane >> 4) << 1;      // 0 or 2
    const int  arow  = lane & 15;             // M within tile
    const bool hi    = (lane >= 16);          // picks (k+2,k+3) from b128
    const bool col0  = (arow == 0);           // lanes 0/16 feed B column 0

    for (int t = 0; t < T_STEPS; ++t) {
        sh_z[tid] = z_seq[((size_t)t * BATCH + b) * DG + tid];
        __syncthreads();

        // ---- base = W_h h_prev + W_g z + b_h  (fp32 WMMA, column-0 matvec)
#pragma unroll 1
        for (int r = 0; r < 4; ++r) {
            const int i0 = (wave + 8 * r) << 4;   // output tile base (16 rows)
            v8f acc0 = {0.f, 0.f, 0.f, 0.f, 0.f, 0.f, 0.f, 0.f};
            v8f acc1 = {0.f, 0.f, 0.f, 0.f, 0.f, 0.f, 0.f, 0.f};

            const float* wrow = Wh + (size_t)(i0 + arow) * DH;
#pragma unroll 1
            for (int kb = 0; kb < DH; kb += 32) {
                __builtin_prefetch(wrow + kb + 96, 0, 3);  // global_prefetch_b8
#pragma unroll
                for (int kk = 0; kk < 32; kk += 8) {
                    const int k = kb + kk;
                    const float4 w0 = *(const float4*)(wrow + k);
                    const float2 h0 = *(const float2*)(sh_h + k + khalf);
                    v2f av, bv;
                    av.x = hi ? w0.z : w0.x;  av.y = hi ? w0.w : w0.y;
                    bv.x = col0 ? h0.x : 0.f; bv.y = col0 ? h0.y : 0.f;
                    acc0 = __builtin_amdgcn_wmma_f32_16x16x4_f32(
                            false, av, false, bv, (short)0, acc0, false, false);
                    const float4 w1 = *(const float4*)(wrow + k + 4);
                    const float2 h1 = *(const float2*)(sh_h + k + 4 + khalf);
                    av.x = hi ? w1.z : w1.x;  av.y = hi ? w1.w : w1.y;
                    bv.x = col0 ? h1.x : 0.f; bv.y = col0 ? h1.y : 0.f;
                    acc1 = __builtin_amdgcn_wmma_f32_16x16x4_f32(
                            false, av, false, bv, (short)0, acc1, false, false);
                }
            }
            const float* grow = Wg + (size_t)(i0 + arow) * DG;
#pragma unroll 1
            for (int kb = 0; kb < DG; kb += 32) {
                __builtin_prefetch(grow + kb + 96, 0, 3);
#pragma unroll
                for (int kk = 0; kk < 32; kk += 8) {
                    const int k = kb + kk;
                    const float4 w0 = *(const float4*)(grow + k);
                    const float2 z0 = *(const float2*)(sh_z + k + khalf);
                    v2f av, bv;
                    av.x = hi ? w0.z : w0.x;  av.y = hi ? w0.w : w0.y;
                    bv.x = col0 ? z0.x : 0.f; bv.y = col0 ? z0.y : 0.f;
                    acc0 = __builtin_amdgcn_wmma_f32_16x16x4_f32(
                            false, av, false, bv, (short)0, acc0, false, false);
                    const float4 w1 = *(const float4*)(grow + k + 4);
                    const float2 z1 = *(const float2*)(sh_z + k + 4 + khalf);
                    av.x = hi ? w1.z : w1.x;  av.y = hi ? w1.w : w1.y;
                    bv.x = col0 ? z1.x : 0.f; bv.y = col0 ? z1.y : 0.f;
                    acc1 = __builtin_amdgcn_wmma_f32_16x16x4_f32(
                            false, av, false, bv, (short)0, acc1, false, false);
                }
            }
            // D column 0 lives in lane 0 (M=0..7) and lane 16 (M=8..15).
            if (col0) {
                const int mb = i0 + ((lane >> 4) << 3);
#pragma unroll
                for (int m = 0; m < 8; ++m)
                    sh_base[mb + m] = acc0[m] + acc1[m] + sh_bh[mb + m];
            }
        }
        __syncthreads();

        // ---- h = relu(LN(base))
        ln_relu(sh_base[tid], sh_base[tid + 256], sh_g, sh_bt, sh_h, sh_red, tid);
        __syncthreads();

        // ---- 3 inner settling iterations: A.h via decayed history
        for (int it = 0; it < 3; ++it) {
            // d[s] = h_s . h   (each wave owns s = wave, wave+8, ...)
            for (int s = wave; s < t; s += 8) {
                const float* hr = sh_hist + s * DH;
                float p = 0.f;
#pragma unroll 4
                for (int j = lane; j < DH; j += 32) p += hr[j] * sh_h[j];
                p = wave_sum(p);
                if (lane == 0) sh_d[s] = p;
            }
            __syncthreads();
            if (tid < t) sh_d[tid] = ETA * sh_pow[t - 1 - tid] * sh_d[tid];
            __syncthreads();
            // tmp = base + sum_s coef[s] * h_s ; then LN+ReLU
            float a0 = sh_base[tid], a1 = sh_base[tid + 256];
            for (int s = 0; s < t; ++s) {
                const float c = sh_d[s];
                a0 += c * sh_hist[s * DH + tid];
                a1 += c * sh_hist[s * DH + tid + 256];
            }
            ln_relu(a0, a1, sh_g, sh_bt, sh_h, sh_red, tid);
            __syncthreads();
        }

        // ---- record h_t (implicitly applies A <- lam*A + eta*h h^T)
        sh_hist[t * DH + tid]       = sh_h[tid];
        sh_hist[t * DH + tid + 256] = sh_h[tid + 256];
        __syncthreads();
    }

    // ---- head: pred[g] = W_head[g,:] . h + b_head[g]   (g = tid, 0..255)
    const float* wr = Whead + (size_t)tid * DH;
    float pred = bhead[tid];
#pragma unroll 8
    for (int i = 0; i < DH; ++i) pred += sh_h[i] * wr[i];

    // ---- loss / acc contributions for this batch element
    const float cv = clean[(size_t)b * DG + tid];
    float e  = (pred - cv) * (pred - cv);
    float cc = cv * cv;
    float pp = pred * pred;
    float pc = pred * cv;
    e  = wave_sum(e);  cc = wave_sum(cc);
    pp = wave_sum(pp); pc = wave_sum(pc);
    __syncthreads();
    if (lane == 0) {
        sh_red[wave]      = e;  sh_red[8 + wave]  = cc;
        sh_red[16 + wave] = pp; sh_red[24 + wave] = pc;
    }
    __syncthreads();
    if (tid == 0) {
        float E = 0.f, C = 0.f, P = 0.f, PC = 0.f;
        for (int w = 0; w < 8; ++w) {
            E += sh_red[w];      C  += sh_red[8 + w];
            P += sh_red[16 + w]; PC += sh_red[24 + w];
        }
        const float loss_b = E / (C + 1e-6f);
        const float acc_b  = PC / ((sqrtf(P) + 1e-6f) * (sqrtf(C) + 1e-6f));
        atomicAdd(&out[0], loss_b * (1.0f / BATCH));
        atomicAdd(&out[1], acc_b  * (1.0f / BATCH));
    }
}

extern "C" void kernel_launch(void* const* d_in, const int* in_sizes, int n_in,
                              void* d_out, int out_size, void* d_ws, size_t ws_size,
                              hipStream_t stream) {
    (void)in_sizes; (void)n_in; (void)d_ws; (void)ws_size; (void)out_size;
    const float* z_seq = (const float*)d_in[0];
    const float* clean = (const float*)d_in[1];
    const float* Wh    = (const float*)d_in[2];
    const float* Wg    = (const float*)d_in[3];
    const float* bh    = (const float*)d_in[4];
    const float* gamma = (const float*)d_in[5];
    const float* beta  = (const float*)d_in[6];
    const float* Whead = (const float*)d_in[7];
    const float* bhead = (const float*)d_in[8];
    float* out = (float*)d_out;

    init_out_kernel<<<1, 32, 0, stream>>>(out);
    fastweight_rnn_kernel<<<BATCH, 256, 0, stream>>>(
        z_seq, clean, Wh, Wg, bh, gamma, beta, Whead, bhead, out);
}